// DNN_WPE_2216203125305
// MI455X (gfx1250) — compile-verified
//
#include <hip/hip_runtime.h>

typedef __attribute__((ext_vector_type(2))) float v2f;
typedef __attribute__((ext_vector_type(8))) float v8f;

// Problem constants (fixed by reference setup_inputs)
constexpr int B = 4, T = 512, C = 8, F = 257;
constexpr int TAPS = 5, DELAY = 3;
constexpr int KC  = TAPS * C;              // 40 stacked-tap rows
constexpr int NR  = 48;                    // padded rows; cols = 40 (R) + 8 (rhs)
constexpr int MS  = 50;                    // LDS row stride for M (conflict pad)
constexpr int TP  = 520;                   // padded time extent in LDS (16B-mult)
constexpr int KF  = T - DELAY - TAPS + 1;  // 505 frames
constexpr int KCH = 127;                   // ceil(505/4) K-chunks of 4
constexpr float EPS     = 1e-6f;
constexpr float WPE_EPS = 1e-10f;
constexpr size_t NO = (size_t)B * T * C * F;   // one output plane
constexpr size_t NP = (size_t)B * F * C * T;   // transposed plane

// ---------------------------------------------------------------------------
// 1) (B,T,C,F) -> (B,F,C,T) planes via LDS-tiled transpose
// ---------------------------------------------------------------------------
__global__ __launch_bounds__(256)
void wpe_transpose(const float* __restrict__ inr, const float* __restrict__ ini,
                   float* __restrict__ yr, float* __restrict__ yi) {
  __shared__ float tr[32][33];
  __shared__ float ti[32][33];
  const int tx = threadIdx.x, ty = threadIdx.y;
  const int f0 = blockIdx.x * 32, t0 = blockIdx.y * 32;
  const int b = blockIdx.z >> 3, c = blockIdx.z & 7;
#pragma unroll
  for (int j = 0; j < 4; ++j) {
    int t = t0 + ty + 8 * j;
    int f = f0 + tx;
    float vr = 0.f, vi = 0.f;
    if (f < F) {
      size_t idx = ((size_t)(b * T + t) * C + c) * F + f;
      vr = inr[idx]; vi = ini[idx];
    }
    tr[ty + 8 * j][tx] = vr;
    ti[ty + 8 * j][tx] = vi;
  }
  __syncthreads();
#pragma unroll
  for (int j = 0; j < 4; ++j) {
    int f = f0 + ty + 8 * j;
    int t = t0 + tx;
    if (f < F) {
      size_t idx = ((size_t)(b * F + f) * C + c) * T + t;
      yr[idx] = tr[tx][ty + 8 * j];
      yi[idx] = ti[tx][ty + 8 * j];
    }
  }
}

// ---------------------------------------------------------------------------
// 2) power(b,f,t) = max(mean_c |y|^2, EPS)  -> written into d_out section 3
// ---------------------------------------------------------------------------
__global__ __launch_bounds__(256)
void wpe_power(const float* __restrict__ yr, const float* __restrict__ yi,
               float* __restrict__ power) {
  int idx = blockIdx.x * 256 + threadIdx.x;
  if (idx >= B * F * T) return;
  int t  = idx % T;
  int bf = idx / T;
  float s = 0.f;
#pragma unroll
  for (int c = 0; c < C; ++c) {
    size_t i = ((size_t)bf * C + c) * T + t;
    float a = yr[i], bb = yi[i];
    s += a * a + bb * bb;
  }
  s *= (1.0f / (float)C);
  power[idx] = fmaxf(s, EPS);
}

// ---------------------------------------------------------------------------
// 3) Per-(b,f): WMMA gram  M = Psi_w^H [Psi | Ytgt]  (48x48 complex),
//    then in-LDS Gauss-Jordan solve of (R + lam I) G = v^T, write G.
//    128 threads = 4 wave32; 9 output tiles of 16x16 over the waves.
//    Y staged into LDS with global_load_async_to_lds_b128 (ASYNCcnt).
// ---------------------------------------------------------------------------
__global__ __launch_bounds__(128)
void wpe_solve(const float* __restrict__ yr, const float* __restrict__ yi,
               const float* __restrict__ power, float2* __restrict__ G2) {
  __shared__ float yrl[C][TP];
  __shared__ float yil[C][TP];
  __shared__ float wl[TP];
  __shared__ float Mr[NR * MS];
  __shared__ float Mi[NR * MS];
  __shared__ float frr[KC], fri[KC];
  __shared__ float sLam, sIpr, sIpi;

  const int tid = threadIdx.x;
  const int bf  = blockIdx.x;           // b*F + f
  const int b   = bf / F;
  const int f   = bf - b * F;

  // ---- Async DMA: global planes -> LDS, 16B per lane per op --------------
  {
    const float* baseR = yr + (size_t)bf * C * T;
    const float* baseI = yi + (size_t)bf * C * T;
    unsigned ldsR = (unsigned)(uintptr_t)&yrl[0][0];
    unsigned ldsI = (unsigned)(uintptr_t)&yil[0][0];
    for (int i = tid; i < C * (T / 4); i += 128) {
      int d  = i >> 7;            // / (T/4)
      int t4 = (i & 127) << 2;    // 16B-aligned time offset
      unsigned lR = ldsR + (unsigned)((d * TP + t4) * 4);
      unsigned lI = ldsI + (unsigned)((d * TP + t4) * 4);
      const float* gR = baseR + d * T + t4;
      const float* gI = baseI + d * T + t4;
      asm volatile("global_load_async_to_lds_b128 %0, %1, off"
                   :: "v"(lR), "v"(gR) : "memory");
      asm volatile("global_load_async_to_lds_b128 %0, %1, off"
                   :: "v"(lI), "v"(gI) : "memory");
    }
  }
  // Zero-pad t in [T, TP) (regular DS path, ordered by the barrier).
  if (tid < C * (TP - T)) {
    int d = tid / (TP - T);
    int t = T + tid % (TP - T);
    yrl[d][t] = 0.f;
    yil[d][t] = 0.f;
  }
  // Frame weights w(t) = 1/max(power(t+delay+taps-1), WPE_EPS); zero pad.
  for (int t = tid; t < TP; t += 128) {
    float w = 0.f;
    if (t < KF) {
      float p = power[(size_t)bf * T + (t + DELAY + TAPS - 1)];
      w = 1.0f / fmaxf(p, WPE_EPS);
    }
    wl[t] = w;
  }
  asm volatile("s_wait_asynccnt 0" ::: "memory");
  __syncthreads();

  // ---- WMMA gram accumulation -------------------------------------------
  const int lane = tid & 31;
  const int half = lane >> 4;
  const int m    = lane & 15;
  const int klo  = half * 2;
  // Wave id as an SGPR so tile-count guards are scalar branches (EXEC stays
  // all-ones through the WMMA region, as the ISA requires).
  const int wave = __builtin_amdgcn_readfirstlane(tid) >> 5;

  // Tile assignment: tiles wave, wave+4, wave+8 (wave0 gets 3, others 2).
  const int nT = (wave == 0) ? 3 : 2;
  int rts[3], cts[3];
  {
    int t0i = wave, t1i = wave + 4, t2i = (wave + 8 < 9) ? wave + 8 : 0;
    rts[0] = t0i / 3; cts[0] = t0i % 3;
    rts[1] = t1i / 3; cts[1] = t1i % 3;
    rts[2] = t2i / 3; cts[2] = t2i % 3;
  }

  float maskA[3];
  int aBase[3], bBase[3];
#pragma unroll
  for (int s = 0; s < 3; ++s) {
    int r  = rts[s] * 16 + m;               // output row (A index)
    int kA = r >> 3; if (kA > TAPS - 1) kA = TAPS - 1;
    int dA = r & 7;
    maskA[s] = (r < KC) ? 1.f : 0.f;
    aBase[s] = dA * TP + (TAPS - 1 - kA);   // Psi row: y[d][t + 4 - k]
    int sc   = cts[s] * 16 + m;             // output col (B index)
    int dB   = sc & 7;
    int offB = (sc < KC) ? (TAPS - 1 - (sc >> 3)) : (DELAY + TAPS - 1);
    bBase[s] = dB * TP + offB;              // cols >=40: target y[e][t+7]
  }

  const float* YR = &yrl[0][0];
  const float* YI = &yil[0][0];

  v8f zero = {};
  v8f accMr[3], accMi[3];
#pragma unroll
  for (int s = 0; s < 3; ++s) { accMr[s] = zero; accMi[s] = zero; }

  int kk = klo;
  for (int ch = 0; ch < KCH; ++ch, kk += 4) {
    float w0 = wl[kk], w1 = wl[kk + 1];
#pragma unroll
    for (int s = 0; s < 3; ++s) {
      if (s < nT) {                          // scalar (SGPR) condition
        int a0 = aBase[s] + kk;
        int b0 = bBase[s] + kk;
        v2f ar, ai, nai, br, bi;
        ar.x = maskA[s] * w0 * YR[a0];
        ar.y = maskA[s] * w1 * YR[a0 + 1];
        ai.x = maskA[s] * w0 * YI[a0];
        ai.y = maskA[s] * w1 * YI[a0 + 1];
        nai.x = -ai.x; nai.y = -ai.y;
        br.x = YR[b0]; br.y = YR[b0 + 1];
        bi.x = YI[b0]; bi.y = YI[b0 + 1];
        // Mr += Ai*Bi + Ar*Br ; Mi += Ar*Bi - Ai*Br  (conj on A side)
        accMr[s] = __builtin_amdgcn_wmma_f32_16x16x4_f32(
            false, ai, false, bi, (short)0, accMr[s], false, false);
        accMr[s] = __builtin_amdgcn_wmma_f32_16x16x4_f32(
            false, ar, false, br, (short)0, accMr[s], false, false);
        accMi[s] = __builtin_amdgcn_wmma_f32_16x16x4_f32(
            false, ar, false, bi, (short)0, accMi[s], false, false);
        accMi[s] = __builtin_amdgcn_wmma_f32_16x16x4_f32(
            false, nai, false, br, (short)0, accMi[s], false, false);
      }
    }
  }

  // Spill D-fragments to LDS M (row = rt*16 + i + 8*half, col = ct*16 + m).
#pragma unroll
  for (int s = 0; s < 3; ++s) {
    if (s < nT) {
      int row0 = rts[s] * 16 + 8 * half;
      int col  = cts[s] * 16 + m;
#pragma unroll
      for (int i = 0; i < 8; ++i) {
        Mr[(row0 + i) * MS + col] = accMr[s][i];
        Mi[(row0 + i) * MS + col] = accMi[s][i];
      }
    }
  }
  __syncthreads();

  // ---- Diagonal loading: lam = WPE_EPS * trace(R).real / 40 --------------
  if (tid == 0) {
    float trc = 0.f;
    for (int p = 0; p < KC; ++p) trc += Mr[p * MS + p];
    sLam = trc * (WPE_EPS / (float)KC);
  }
  __syncthreads();
  if (tid < KC) Mr[tid * MS + tid] += sLam;
  __syncthreads();

  // ---- Gauss-Jordan on augmented 40x48 complex system --------------------
  for (int p = 0; p < KC; ++p) {
    if (tid == 0) {
      float pr = Mr[p * MS + p], pi = Mi[p * MS + p];
      float den = pr * pr + pi * pi + 1e-30f;
      sIpr = pr / den; sIpi = -pi / den;
    }
    __syncthreads();
    if (tid < NR) {                                  // normalize pivot row
      float xr = Mr[p * MS + tid], xi = Mi[p * MS + tid];
      Mr[p * MS + tid] = xr * sIpr - xi * sIpi;
      Mi[p * MS + tid] = xr * sIpi + xi * sIpr;
    }
    __syncthreads();
    if (tid < KC) {                                  // snapshot column p
      frr[tid] = Mr[tid * MS + p];
      fri[tid] = Mi[tid * MS + p];
    }
    __syncthreads();
    for (int idx = tid; idx < (KC - 1) * NR; idx += 128) {
      int r = idx / NR;
      int s = idx - r * NR;
      r += (r >= p) ? 1 : 0;                         // skip pivot row
      float fr = frr[r], fi = fri[r];
      float pr = Mr[p * MS + s], pi2 = Mi[p * MS + s];
      Mr[r * MS + s] -= fr * pr - fi * pi2;
      Mi[r * MS + s] -= fr * pi2 + fi * pr;
    }
    __syncthreads();
  }

  // G[r][e] now in cols 40..47; store f-contiguous: G2[((b*40+r)*8+e)*F + f]
  for (int i = tid; i < KC * C; i += 128) {
    int r = i >> 3, e = i & 7;
    float2 g;
    g.x = Mr[r * MS + KC + e];
    g.y = Mi[r * MS + KC + e];
    G2[((size_t)(b * KC + r) * C + e) * F + f] = g;
  }
}

// ---------------------------------------------------------------------------
// 4) enh = Y - sum_{tau,d} G[tau,d,e] * Y[d][t-delay-tau], ilens mask,
//    write (B,T,C,F). Fully coalesced over f; G streams from L2 (2.6 MB).
// ---------------------------------------------------------------------------
__global__ __launch_bounds__(256)
void wpe_tail(const float* __restrict__ inr, const float* __restrict__ ini,
              const float2* __restrict__ G2, const int* __restrict__ ilens,
              float* __restrict__ outr, float* __restrict__ outi) {
  int f = blockIdx.x * 256 + threadIdx.x;
  if (f >= F) return;
  int t = blockIdx.y;
  int b = blockIdx.z;
  int len = ilens[b];
  size_t base = ((size_t)(b * T + t) * C) * F + f;
  if (t >= len) {
#pragma unroll
    for (int e = 0; e < C; ++e) {
      outr[base + (size_t)e * F] = 0.f;
      outi[base + (size_t)e * F] = 0.f;
    }
    return;
  }
  float accR[C], accI[C];
#pragma unroll
  for (int e = 0; e < C; ++e) {
    accR[e] = inr[base + (size_t)e * F];
    accI[e] = ini[base + (size_t)e * F];
  }
#pragma unroll
  for (int tau = 0; tau < TAPS; ++tau) {
    int s = DELAY + tau;
    if (t >= s) {
      size_t yb = ((size_t)(b * T + (t - s)) * C) * F + f;
#pragma unroll
      for (int d = 0; d < C; ++d) {
        float ydr = inr[yb + (size_t)d * F];
        float ydi = ini[yb + (size_t)d * F];
        size_t gb = ((size_t)(b * KC + tau * C + d) * C) * F + f;
#pragma unroll
        for (int e = 0; e < C; ++e) {
          float2 g = G2[gb + (size_t)e * F];
          accR[e] -= g.x * ydr - g.y * ydi;
          accI[e] -= g.x * ydi + g.y * ydr;
        }
      }
    }
  }
#pragma unroll
  for (int e = 0; e < C; ++e) {
    outr[base + (size_t)e * F] = accR[e];
    outi[base + (size_t)e * F] = accI[e];
  }
}

// ---------------------------------------------------------------------------
extern "C" void kernel_launch(void* const* d_in, const int* in_sizes, int n_in,
                              void* d_out, int out_size, void* d_ws, size_t ws_size,
                              hipStream_t stream) {
  const float* inr   = (const float*)d_in[0];
  const float* ini   = (const float*)d_in[1];
  const int*   ilens = (const int*)d_in[2];

  float* out   = (float*)d_out;
  float* outr  = out;                 // enh.real (B,T,C,F)
  float* outi  = out + NO;            // enh.imag (B,T,C,F)
  float* power = out + 2 * NO;        // power    (B,F,T)

  float*  ws = (float*)d_ws;
  float*  yr = ws;                    // (B,F,C,T)
  float*  yi = ws + NP;
  float2* G2 = (float2*)(ws + 2 * NP);  // (B,40,8,F) complex

  dim3 tb(32, 8);
  dim3 tg((F + 31) / 32, T / 32, B * C);
  wpe_transpose<<<tg, tb, 0, stream>>>(inr, ini, yr, yi);

  int ptot = B * F * T;
  wpe_power<<<(ptot + 255) / 256, 256, 0, stream>>>(yr, yi, power);

  wpe_solve<<<B * F, 128, 0, stream>>>(yr, yi, power, G2);

  dim3 tg3((F + 255) / 256, T, B);
  wpe_tail<<<tg3, 256, 0, stream>>>(inr, ini, G2, ilens, outr, outi);
}